// BatchelorGPUNUFFTFwd_43301860278795
// MI455X (gfx1250) — compile-verified
//
#include <hip/hip_runtime.h>
#include <hip/hip_bf16.h>
#include <math.h>

// Problem constants (match reference)
constexpr int NX = 128, NY = 128, NC = 8, NS = 2048, NT = 4;

typedef __attribute__((ext_vector_type(2))) float v2f;
typedef __attribute__((ext_vector_type(8))) float v8f;

// Workspace plane sizes (floats)
constexpr size_t GA_PLANE  = (size_t)NT * NC * NY * NX;   // 524288 (2 MB)
constexpr size_t EY_PLANE  = (size_t)NT * NY * NS;        // 1048576 (4 MB)
constexpr size_t EX_ELEMS  = (size_t)NT * NS * NX;        // float2 elements (8 MB)

// ---------------------------------------------------------------------------
// Kernel 1: bilinear warp of complex image + coil-map multiply.
// Writes split planes with K(y)-pairs innermost:
//   gAre[t][c][y/2][x][y&1], gAim[...]  -> WMMA A fragment = one b64 load.
// ---------------------------------------------------------------------------
__global__ void warp_csm_kernel(const float* __restrict__ img_re,
                                const float* __restrict__ img_im,
                                const float* __restrict__ csm_re,
                                const float* __restrict__ csm_im,
                                const float* __restrict__ flow,
                                float* __restrict__ gAre,
                                float* __restrict__ gAim) {
    int idx = blockIdx.x * blockDim.x + threadIdx.x;   // NT*NX*NY threads
    if (idx >= NT * NX * NY) return;
    int t = idx / (NX * NY);
    int p = idx % (NX * NY);
    int x = p % NX;
    int y = p / NX;

    // flow layout: (NX, NY, 2, NT)
    float fx = flow[((x * NY + y) * 2 + 0) * NT + t];
    float fy = flow[((x * NY + y) * 2 + 1) * NT + t];
    float gx = (float)x + fx;
    float gy = (float)y + fy;
    float x0f = floorf(gx), y0f = floorf(gy);
    float wx = gx - x0f, wy = gy - y0f;
    int x0 = (int)x0f, y0 = (int)y0f;

    float wr = 0.f, wi = 0.f;
    #pragma unroll
    for (int dx = 0; dx < 2; ++dx) {
        #pragma unroll
        for (int dy = 0; dy < 2; ++dy) {
            int xi = x0 + dx, yi = y0 + dy;
            float w = (dx ? wx : (1.f - wx)) * (dy ? wy : (1.f - wy));
            bool valid = (xi >= 0) & (xi < NX) & (yi >= 0) & (yi < NY);
            int xc = min(max(xi, 0), NX - 1);
            int yc = min(max(yi, 0), NY - 1);
            float wv = valid ? w : 0.f;
            wr += wv * img_re[xc * NY + yc];
            wi += wv * img_im[xc * NY + yc];
        }
    }

    int ypair = y >> 1, ybit = y & 1;
    #pragma unroll
    for (int c = 0; c < NC; ++c) {
        float cr = csm_re[(c * NX + x) * NY + y];
        float ci = csm_im[(c * NX + x) * NY + y];
        float gr = cr * wr - ci * wi;
        float gi = cr * wi + ci * wr;
        size_t o = (((size_t)(t * NC + c) * (NY / 2) + ypair) * NX + x) * 2 + ybit;
        gAre[o] = gr;
        gAim[o] = gi;
    }
}

// ---------------------------------------------------------------------------
// Kernel 2: phase tables.
//   eyB{re,im}[t][y/2][s][y&1] = cos/sin(-2*pi * ky[s,t] * (y-64))
//        -> WMMA B fragment = one b64 load, lane-coalesced over s.
//   Ex[t][s][x] = exp(-2*pi*i * kx[s,t] * (x-64))  (interleaved, VALU stage)
// ---------------------------------------------------------------------------
__global__ void phase_kernel(const float* __restrict__ traj,
                             float* __restrict__ eyBre,
                             float* __restrict__ eyBim,
                             float2* __restrict__ Ex) {
    constexpr float TWO_PI = 6.283185307179586f;
    int idx = blockIdx.x * blockDim.x + threadIdx.x;   // 2*NT*NY*NS threads
    int half = NT * NY * NS;
    if (idx < half) {
        int s = idx % NS;
        int y = (idx / NS) % NY;
        int t = idx / (NS * NY);
        float ky = traj[(s * 2 + 1) * NT + t];         // traj: (NS,2,NT)
        float ph = -TWO_PI * ky * (float)(y - NY / 2);
        float si, co;
        __sincosf(ph, &si, &co);
        size_t o = (((size_t)t * (NY / 2) + (y >> 1)) * NS + s) * 2 + (y & 1);
        eyBre[o] = co;
        eyBim[o] = si;
    } else if (idx < 2 * half) {
        int j = idx - half;
        int x = j % NX;
        int s = (j / NX) % NS;
        int t = j / (NX * NS);
        float kx = traj[(s * 2 + 0) * NT + t];
        float ph = -TWO_PI * kx * (float)(x - NX / 2);
        float si, co;
        __sincosf(ph, &si, &co);
        Ex[((size_t)t * NS + s) * NX + x] = make_float2(co, si);
    }
}

// ---------------------------------------------------------------------------
// Kernel 3: zero the output accumulator.
// ---------------------------------------------------------------------------
__global__ void zero_kernel(float* __restrict__ out, int n) {
    int i = blockIdx.x * blockDim.x + threadIdx.x;
    if (i < n) out[i] = 0.f;
}

// ---------------------------------------------------------------------------
// Kernel 4: complex GEMM via V_WMMA_F32_16X16X4_F32 + fused Ex contraction.
// Each of Ar/Ai/Br/Bi is a single b64 load straight into an even VGPR pair
// (no deinterleave moves). 4 WMMAs (RR,II,RI,IR) per K=4 step, K=128, 8
// x-blocks per wave; Stage-2 Ex contraction in VALU; frame-sum via atomics.
// ---------------------------------------------------------------------------
__global__ void __launch_bounds__(256)
nufft_wmma_kernel(const float* __restrict__ gAre,
                  const float* __restrict__ gAim,
                  const float* __restrict__ eyBre,
                  const float* __restrict__ eyBim,
                  const float2* __restrict__ Ex,
                  float* __restrict__ out) {
    int lane = threadIdx.x & 31;
    int wave = threadIdx.x >> 5;                 // 8 waves / block
    int s0 = (blockIdx.x * 8 + wave) * 16;       // s-block base
    int c  = blockIdx.y;
    int t  = blockIdx.z;

    // View planes as v2f arrays: [K/2][row] with K-pair innermost.
    const v2f* Are = (const v2f*)(gAre + (size_t)(t * NC + c) * NY * NX);
    const v2f* Aim = (const v2f*)(gAim + (size_t)(t * NC + c) * NY * NX);
    const v2f* Bre = (const v2f*)(eyBre + (size_t)t * NY * NS);
    const v2f* Bim = (const v2f*)(eyBim + (size_t)t * NY * NS);
    const float2* Ex_t = Ex + (size_t)t * NS * NX;

    int m  = lane & 15;              // A row (x) / B col (s) within tile
    int kp = (lane >> 4);            // upper half-wave handles K-pair +1

    float outRe = 0.f, outIm = 0.f;

    for (int xb = 0; xb < 8; ++xb) {
        int x0 = xb * 16;
        v8f accRR = {}, accII = {}, accRI = {}, accIR = {};

        const v2f* arp = Are + (size_t)kp * NX + x0 + m;
        const v2f* aip = Aim + (size_t)kp * NX + x0 + m;
        const v2f* brp = Bre + (size_t)kp * NS + s0 + m;
        const v2f* bip = Bim + (size_t)kp * NS + s0 + m;

        #pragma unroll 4
        for (int k = 0; k < NY / 4; ++k) {       // each step covers K=4
            v2f Ar = arp[(size_t)2 * k * NX];
            v2f Ai = aip[(size_t)2 * k * NX];
            v2f Br = brp[(size_t)2 * k * NS];
            v2f Bi = bip[(size_t)2 * k * NS];
            accRR = __builtin_amdgcn_wmma_f32_16x16x4_f32(
                false, Ar, false, Br, (short)0, accRR, false, false);
            accII = __builtin_amdgcn_wmma_f32_16x16x4_f32(
                false, Ai, false, Bi, (short)0, accII, false, false);
            accRI = __builtin_amdgcn_wmma_f32_16x16x4_f32(
                false, Ar, false, Bi, (short)0, accRI, false, false);
            accIR = __builtin_amdgcn_wmma_f32_16x16x4_f32(
                false, Ai, false, Br, (short)0, accIR, false, false);
        }

        // Stage 2: out[c,s] += sum_x tmp[x,s] * Ex[s,x]
        // Lane holds D rows M=r (lanes<16) or M=r+8 (lanes>=16), col N=m.
        int s = s0 + m;
        int xbase = x0 + ((lane >> 4) << 3);
        const float2* ex = Ex_t + (size_t)s * NX + xbase;   // 8 float2
        #pragma unroll
        for (int r = 0; r < 8; ++r) {
            float tr = accRR[r] - accII[r];
            float ti = accRI[r] + accIR[r];
            float2 e = ex[r];
            outRe += tr * e.x - ti * e.y;
            outIm += tr * e.y + ti * e.x;
        }
    }

    // Combine the two half-waves (x 0..7 and 8..15 partials share the same s).
    outRe += __shfl_xor(outRe, 16, 32);
    outIm += __shfl_xor(outIm, 16, 32);

    int s = s0 + m;
    if (lane < 16) atomicAdd(&out[0 * NC * NS + c * NS + s], outRe);
    else           atomicAdd(&out[1 * NC * NS + c * NS + s], outIm);
}

// ---------------------------------------------------------------------------
extern "C" void kernel_launch(void* const* d_in, const int* in_sizes, int n_in,
                              void* d_out, int out_size, void* d_ws, size_t ws_size,
                              hipStream_t stream) {
    const float* img_re = (const float*)d_in[0];
    const float* img_im = (const float*)d_in[1];
    const float* csm_re = (const float*)d_in[2];
    const float* csm_im = (const float*)d_in[3];
    const float* flow   = (const float*)d_in[4];
    const float* traj   = (const float*)d_in[5];
    // d_in[6] = dcf (unused in forward op)
    float* out = (float*)d_out;

    float* ws = (float*)d_ws;
    float*  gAre  = ws;
    float*  gAim  = gAre  + GA_PLANE;
    float*  eyBre = gAim  + GA_PLANE;
    float*  eyBim = eyBre + EY_PLANE;
    float2* Ex    = (float2*)(eyBim + EY_PLANE);

    // 1) warp + coil maps -> split-plane A operands
    warp_csm_kernel<<<(NT * NX * NY + 255) / 256, 256, 0, stream>>>(
        img_re, img_im, csm_re, csm_im, flow, gAre, gAim);

    // 2) phase tables -> split-plane B operands + interleaved Ex
    phase_kernel<<<(2 * NT * NY * NS + 255) / 256, 256, 0, stream>>>(
        traj, eyBre, eyBim, Ex);

    // 3) zero output (frame sum accumulates via atomics)
    zero_kernel<<<(2 * NC * NS + 255) / 256, 256, 0, stream>>>(out, 2 * NC * NS);

    // 4) WMMA NUFFT: grid = (s-blocks/8, coils, frames), 8 waves/block
    dim3 grid(NS / (16 * 8), NC, NT);
    nufft_wmma_kernel<<<grid, 256, 0, stream>>>(gAre, gAim, eyBre, eyBim, Ex, out);
}